// MultiHeadAttentionSum_25555055411781
// MI455X (gfx1250) — compile-verified
//
#include <hip/hip_runtime.h>

typedef __attribute__((ext_vector_type(2))) float v2f;
typedef __attribute__((ext_vector_type(8))) float v8f;

#define L_SEQ 1024
#define BSZ   8
#define EMB   512
#define NH    8
#define DQK   64
#define OCH   1536            // 3*NH*DQK
#define MROWS 8192            // L_SEQ*BSZ
#define SCALING 0.125f        // (EMB/NH)^-0.5
#define GN_EPS 1e-5f

__device__ __forceinline__ v8f wmma_f32_k4(v2f a, v2f b, v8f c) {
  // V_WMMA_F32_16X16X4_F32 : D = A(16x4) * B(4x16) + C(16x16)
  return __builtin_amdgcn_wmma_f32_16x16x4_f32(false, a, false, b,
                                               (short)0, c, false, false);
}

// ---------------------------------------------------------------------------
// Kernel 0: zero the attention-weight accumulator (64 * 1024 floats)
// ---------------------------------------------------------------------------
__global__ void zero_aw_kernel(float* __restrict__ aw) {
  aw[blockIdx.x * 256 + threadIdx.x] = 0.0f;
}

// ---------------------------------------------------------------------------
// Kernel 1: ft[m,o] = emb_flat[m,:] . W[o,:] + b[o]
//   m = l*BSZ + bs  (emb is [L, BSZ, EMB] row-major)
//   Each wave: 16 rows x 64 cols, fp32 WMMA, K stepped by 4.
// ---------------------------------------------------------------------------
__global__ __launch_bounds__(256) void qkv_gemm_kernel(
    const float* __restrict__ A,   // [MROWS, EMB]
    const float* __restrict__ W,   // [OCH, EMB]
    const float* __restrict__ bias,// [OCH]
    float* __restrict__ ft) {      // [MROWS, OCH]
  const int lane    = threadIdx.x & 31;
  const int wave    = threadIdx.x >> 5;            // 0..7
  const int rowBase = blockIdx.x * 128 + wave * 16;
  const int colBase = blockIdx.y * 64;
  const int laneM   = lane & 15;
  const int koff    = (lane >> 4) * 2;             // K offset per A/B layout

  v8f c0 = {}, c1 = {}, c2 = {}, c3 = {};
  const float* arow = A + (size_t)(rowBase + laneM) * EMB + koff;
  const float* b0 = W + (size_t)(colBase +  0 + laneM) * EMB + koff;
  const float* b1 = W + (size_t)(colBase + 16 + laneM) * EMB + koff;
  const float* b2 = W + (size_t)(colBase + 32 + laneM) * EMB + koff;
  const float* b3 = W + (size_t)(colBase + 48 + laneM) * EMB + koff;

  for (int k0 = 0; k0 < EMB; k0 += 4) {
    v2f a = *(const v2f*)(arow + k0);
    c0 = wmma_f32_k4(a, *(const v2f*)(b0 + k0), c0);
    c1 = wmma_f32_k4(a, *(const v2f*)(b1 + k0), c1);
    c2 = wmma_f32_k4(a, *(const v2f*)(b2 + k0), c2);
    c3 = wmma_f32_k4(a, *(const v2f*)(b3 + k0), c3);
  }

  const int mHalf = (lane >> 4) * 8;
  const int nLane = lane & 15;
#pragma unroll
  for (int r = 0; r < 8; ++r) {
    const int m = rowBase + mHalf + r;
    float* dst = ft + (size_t)m * OCH + colBase + nLane;
    dst[0]  = c0[r] + bias[colBase +       nLane];
    dst[16] = c1[r] + bias[colBase + 16 +  nLane];
    dst[32] = c2[r] + bias[colBase + 32 +  nLane];
    dst[48] = c3[r] + bias[colBase + 48 +  nLane];
  }
}

// ---------------------------------------------------------------------------
// Kernel 2: per head-batch b' = bs*8 + h:
//   s[l,m] = SCALING * q[l].k[m]; mask = (s==0);
//   w = softmax_row(s) with masked entries excluded;
//   aw[b',m] += sum_l w[l,m]
// One block per (b', 128-query chunk). K staged through LDS in 256-key chunks.
// ---------------------------------------------------------------------------
__global__ __launch_bounds__(256) void attn_colsum_kernel(
    const float* __restrict__ ft,  // [MROWS, OCH]
    float* __restrict__ aw) {      // [64, L_SEQ]
  __shared__ float ksh[256 * DQK];               // 64 KB

  const int bprime = blockIdx.x;                 // 0..63
  const int lchunk = blockIdx.y;                 // 0..7
  const int bs   = bprime >> 3;
  const int h    = bprime & 7;
  const int lane  = threadIdx.x & 31;
  const int wave  = threadIdx.x >> 5;
  const int laneM = lane & 15;
  const int koff  = (lane >> 4) * 2;
  const int qch = h * 192;                       // q channel base in OCH
  const int kch = qch + 64;                      // k channel base

  // Preload this wave's 16 query rows as 16 A-fragments (d = 0..63)
  const int l = lchunk * 128 + wave * 16 + laneM;
  const float* qrow = ft + (size_t)(l * BSZ + bs) * OCH + qch + koff;
  v2f afr[16];
#pragma unroll
  for (int kk = 0; kk < 16; ++kk) afr[kk] = *(const v2f*)(qrow + kk * 4);

  // ---------------- pass 1: row sums of exp ----------------
  v8f rowacc = {};
  for (int mc = 0; mc < 4; ++mc) {
    __syncthreads();
    for (int i = threadIdx.x; i < 256 * (DQK / 4); i += 256) {
      const int krow = i >> 4;                   // 0..255
      const int seg  = i & 15;                   // 4 floats each
      const float4 v = *(const float4*)(
          ft + (size_t)((mc * 256 + krow) * BSZ + bs) * OCH + kch + seg * 4);
      *(float4*)(&ksh[krow * DQK + seg * 4]) = v;
    }
    __syncthreads();
    for (int mt = 0; mt < 16; ++mt) {
      v8f s = {};
      const float* kb = &ksh[(mt * 16 + laneM) * DQK + koff];
#pragma unroll
      for (int kk = 0; kk < 16; ++kk)
        s = wmma_f32_k4(afr[kk], *(const v2f*)(kb + kk * 4), s);
#pragma unroll
      for (int r = 0; r < 8; ++r) {
        const float sc = s[r] * SCALING;
        rowacc[r] += (sc == 0.0f) ? 0.0f : __expf(sc);
      }
    }
  }
  // Butterfly-reduce across the 16 lanes of each half (rows r and r+8)
#pragma unroll
  for (int off = 1; off < 16; off <<= 1) {
#pragma unroll
    for (int r = 0; r < 8; ++r) rowacc[r] += __shfl_xor(rowacc[r], off, 32);
  }
  v8f inv;
#pragma unroll
  for (int r = 0; r < 8; ++r) inv[r] = 1.0f / rowacc[r];

  // ---------------- pass 2: column sums of normalized weights ----------------
  for (int mc = 0; mc < 4; ++mc) {
    __syncthreads();
    for (int i = threadIdx.x; i < 256 * (DQK / 4); i += 256) {
      const int krow = i >> 4;
      const int seg  = i & 15;
      const float4 v = *(const float4*)(
          ft + (size_t)((mc * 256 + krow) * BSZ + bs) * OCH + kch + seg * 4);
      *(float4*)(&ksh[krow * DQK + seg * 4]) = v;
    }
    __syncthreads();
    for (int mt = 0; mt < 16; ++mt) {
      v8f s = {};
      const float* kb = &ksh[(mt * 16 + laneM) * DQK + koff];
#pragma unroll
      for (int kk = 0; kk < 16; ++kk)
        s = wmma_f32_k4(afr[kk], *(const v2f*)(kb + kk * 4), s);
      float col = 0.0f;
#pragma unroll
      for (int r = 0; r < 8; ++r) {
        const float sc = s[r] * SCALING;
        col += (sc == 0.0f) ? 0.0f : __expf(sc) * inv[r];
      }
      col += __shfl_xor(col, 16, 32);            // add rows 8..15 half
      if (lane < 16)
        atomicAdd(&aw[bprime * L_SEQ + mc * 256 + mt * 16 + lane], col);
    }
  }
}

// ---------------------------------------------------------------------------
// Kernel 3: attn_output[b',d] = sum_m aw[b',m] * v[b',m,d]; GroupNorm (size-1
// groups) over d; out[bs, h*64+d].
// ---------------------------------------------------------------------------
__global__ __launch_bounds__(64) void combine_gn_kernel(
    const float* __restrict__ ft,
    const float* __restrict__ aw,
    const float* __restrict__ gnw,
    const float* __restrict__ gnb,
    float* __restrict__ out) {
  const int bprime = blockIdx.x;
  const int bs = bprime >> 3, h = bprime & 7;
  const int d  = threadIdx.x;                    // 0..63
  const int vch = h * 192 + 128;
  const float* awb = aw + bprime * L_SEQ;

  float acc = 0.0f;
  for (int m = 0; m < L_SEQ; ++m)
    acc += awb[m] * ft[(size_t)(m * BSZ + bs) * OCH + vch + d];

  __shared__ float red[64], red2[64];
  red[d] = acc; red2[d] = acc * acc;
  __syncthreads();
  for (int s = 32; s > 0; s >>= 1) {
    if (d < s) { red[d] += red[d + s]; red2[d] += red2[d + s]; }
    __syncthreads();
  }
  const float mean = red[0] * (1.0f / 64.0f);
  const float var  = red2[0] * (1.0f / 64.0f) - mean * mean;
  const float y = (acc - mean) * rsqrtf(var + GN_EPS);
  out[bs * 512 + h * 64 + d] = y * gnw[h] + gnb[h];
}

// ---------------------------------------------------------------------------
extern "C" void kernel_launch(void* const* d_in, const int* in_sizes, int n_in,
                              void* d_out, int out_size, void* d_ws, size_t ws_size,
                              hipStream_t stream) {
  const float* emb = (const float*)d_in[0];   // [L, BSZ, EMB]
  const float* W   = (const float*)d_in[1];   // [OCH, EMB]
  const float* b   = (const float*)d_in[2];   // [OCH]
  const float* gnw = (const float*)d_in[3];   // [NH]
  const float* gnb = (const float*)d_in[4];   // [NH]
  float* out = (float*)d_out;                 // [BSZ, 512]

  float* ft = (float*)d_ws;                            // 8192*1536 floats (48 MB)
  float* aw = ft + (size_t)MROWS * OCH;                // 64*1024 floats

  zero_aw_kernel<<<dim3((64 * L_SEQ) / 256), 256, 0, stream>>>(aw);
  qkv_gemm_kernel<<<dim3(MROWS / 128, OCH / 64), 256, 0, stream>>>(emb, W, b, ft);
  attn_colsum_kernel<<<dim3(64, 8), 256, 0, stream>>>(ft, aw);
  combine_gn_kernel<<<64, 64, 0, stream>>>(ft, aw, gnw, gnb, out);
}